// ReconstructionLayer_4793183502592
// MI455X (gfx1250) — compile-verified
//
#include <hip/hip_runtime.h>
#include <hip/hip_bf16.h>
#include <cstddef>

// Problem constants (from reference)
#define B_   16
#define N_   1024
#define C2_  64
#define D_   32
#define H_   64
#define NC_  10
#define TOPK 8
#define EPSQ 1e-11f

// workspace layout per batch (floats): w[0..1023], v[1024..1087], S@1088, c@1089
#define WS_STRIDE 1152

typedef float v2f __attribute__((ext_vector_type(2)));
typedef float v8f __attribute__((ext_vector_type(8)));

// ---------------------------------------------------------------------------
// Per-batch prologue: squash scales, top-8 select, class capsule, 9x64 MLP,
// then the rank-1 reduction terms S, v[64], c and w[n] = relu(fc_n) . v
// ---------------------------------------------------------------------------
__global__ __launch_bounds__(256) void caps_prep_kernel(
    const float* __restrict__ first,   // [B,N,H]
    const float* __restrict__ second,  // [B,C2,D]
    const float* __restrict__ classc,  // [B,NC,D]
    const int*   __restrict__ y,       // [B]
    const float* __restrict__ W1,      // [H,D]
    const float* __restrict__ b1,      // [H]
    float* __restrict__ ws)
{
    const int b = blockIdx.x;
    const int t = threadIdx.x;

    __shared__ float s_scale[C2_];
    __shared__ float s_rs[C2_];
    __shared__ int   s_sel[TOPK];
    __shared__ float s_secin[TOPK + 1][D_];
    __shared__ float s_sec[TOPK + 1][H_];
    __shared__ float s_kscale[TOPK + 1];
    __shared__ float s_v[H_];
    __shared__ float s_red[H_];
    __shared__ float s_rsc;

    // 1) squash norms of second capsules
    if (t < C2_) {
        const float* p = second + ((size_t)b * C2_ + t) * D_;
        float sn = 0.f;
        #pragma unroll
        for (int d = 0; d < D_; ++d) { float x = p[d]; sn += x * x; }
        s_scale[t] = sn / (1.f + sn);
        s_rs[t]    = 1.f / sqrtf(sn + EPSQ);
    }
    // class capsule squash (one thread; tiny)
    if (t == 64) {
        const float* p = classc + ((size_t)b * NC_ + y[b]) * D_;
        float sn = 0.f;
        #pragma unroll
        for (int d = 0; d < D_; ++d) { float x = p[d]; sn += x * x; }
        s_kscale[TOPK] = sn / (1.f + sn);
        s_rsc = 1.f / sqrtf(sn + EPSQ);
    }
    __syncthreads();

    // 2) deterministic top-8 by rank (ties broken by lower index, as lax.top_k)
    if (t < C2_) {
        float sj = s_scale[t];
        int rank = 0;
        for (int i = 0; i < C2_; ++i) {
            float si = s_scale[i];
            rank += (si > sj) || (si == sj && i < t);
        }
        if (rank < TOPK) s_sel[rank] = t;
    }
    __syncthreads();

    // 3) build pre-MLP unit vectors [9][32]
    {
        int k = t >> 5, d = t & 31;       // 256 threads -> k in [0,8)
        int j = s_sel[k];
        s_secin[k][d] = second[((size_t)b * C2_ + j) * D_ + d] * s_rs[j];
        if (d == 0) s_kscale[k] = s_scale[j];
    }
    if (t < D_) {
        s_secin[TOPK][t] = classc[((size_t)b * NC_ + y[b]) * D_ + t] * s_rsc;
    }
    __syncthreads();

    // 4) sec = relu(secin @ W1^T + b1)  [9][64]
    for (int o = t; o < (TOPK + 1) * H_; o += 256) {
        int k = o >> 6, h = o & 63;
        float a = b1[h];
        #pragma unroll
        for (int d = 0; d < D_; ++d) a += s_secin[k][d] * W1[h * D_ + d];
        s_sec[k][h] = fmaxf(a, 0.f);
    }
    __syncthreads();

    // 5) v[h], partial c
    if (t < H_) {
        float vh = 0.f, ch = 0.f;
        #pragma unroll
        for (int k = 0; k < TOPK + 1; ++k) {
            float sk = s_kscale[k], e = s_sec[k][t];
            vh += sk * e;
            ch += sk * e * e;
        }
        s_v[t]   = vh;
        s_red[t] = ch;
    }
    __syncthreads();

    float* wsb = ws + (size_t)b * WS_STRIDE;
    if (t == 0) {
        float c = 0.f, S = 0.f;
        for (int h = 0; h < H_; ++h) c += s_red[h];
        #pragma unroll
        for (int k = 0; k < TOPK + 1; ++k) S += s_kscale[k];
        wsb[1088] = S;
        wsb[1089] = c;
    }
    if (t < H_) wsb[1024 + t] = s_v[t];

    // 6) w[n] = relu(fc_n) . v
    for (int n = t; n < N_; n += 256) {
        const float* p = first + (size_t)b * (N_ * H_) + (size_t)n * H_;
        float a = 0.f;
        #pragma unroll
        for (int h = 0; h < H_; ++h) a += fmaxf(p[h], 0.f) * s_v[h];
        wsb[n] = a;
    }
}

// ---------------------------------------------------------------------------
// Batched Gram + affine epilogue via V_WMMA_F32_16X16X4_F32.
// Block = 256 threads (8 waves), computes a 128x128 output tile.
// Each wave owns a 32x64 sub-tile: 2x4 accumulators of v8f.
// ---------------------------------------------------------------------------
#define TILE 128
#define KCH  32
#define LSTR 36   // padded dword stride: 144B (16B aligned), conflict-free

__global__ __launch_bounds__(256) void caps_gram_kernel(
    const float* __restrict__ first,          // [B,N,H]
    const unsigned char* __restrict__ mask,   // [B,N] bool
    const float* __restrict__ ws,
    float* __restrict__ out)                  // [B,N,N]
{
    const int b       = blockIdx.z;
    const int rowBase = blockIdx.y * TILE;
    const int colBase = blockIdx.x * TILE;

    __shared__ float lA[TILE * LSTR];
    __shared__ float lB[TILE * LSTR];

    const int t    = threadIdx.x;
    const int lane = t & 31;
    const int wave = t >> 5;
    const int wr   = wave & 3;   // 4 wave-rows * 32 rows
    const int wc   = wave >> 2;  // 2 wave-cols * 64 cols
    const int lrow = lane & 15;
    const int koff = (lane >> 4) << 1;  // half-wave selects K pair {0,1} / {2,3}

    v8f acc[2][4] = {};

    const float* Ab = first + (size_t)b * (N_ * H_) + (size_t)rowBase * H_;
    const float* Bb = first + (size_t)b * (N_ * H_) + (size_t)colBase * H_;

    for (int kc = 0; kc < H_; kc += KCH) {
        __syncthreads();
        // global -> LDS with relu fused into the copy (b128 traffic)
        for (int i = t; i < TILE * (KCH / 4); i += 256) {  // 1024 float4 per tile
            int r = i >> 3, q = i & 7;
            float4 va = *(const float4*)(Ab + r * H_ + kc + q * 4);
            float4 vb = *(const float4*)(Bb + r * H_ + kc + q * 4);
            float* da = &lA[r * LSTR + q * 4];
            float* db = &lB[r * LSTR + q * 4];
            da[0] = fmaxf(va.x, 0.f); da[1] = fmaxf(va.y, 0.f);
            da[2] = fmaxf(va.z, 0.f); da[3] = fmaxf(va.w, 0.f);
            db[0] = fmaxf(vb.x, 0.f); db[1] = fmaxf(vb.y, 0.f);
            db[2] = fmaxf(vb.z, 0.f); db[3] = fmaxf(vb.w, 0.f);
        }
        __syncthreads();

        #pragma unroll
        for (int kk = 0; kk < KCH; kk += 4) {
            v2f af[2], bf[4];
            #pragma unroll
            for (int i = 0; i < 2; ++i) {
                int idx = (wr * 32 + i * 16 + lrow) * LSTR + kk + koff;
                af[i].x = lA[idx];
                af[i].y = lA[idx + 1];
            }
            #pragma unroll
            for (int j = 0; j < 4; ++j) {
                int idx = (wc * 64 + j * 16 + lrow) * LSTR + kk + koff;
                bf[j].x = lB[idx];
                bf[j].y = lB[idx + 1];
            }
            #pragma unroll
            for (int i = 0; i < 2; ++i)
                #pragma unroll
                for (int j = 0; j < 4; ++j)
                    acc[i][j] = __builtin_amdgcn_wmma_f32_16x16x4_f32(
                        false, af[i], false, bf[j], (short)0, acc[i][j], false, false);
        }
    }

    // epilogue: adj = S*g + w[row] + w[col] + c, row-masked
    const float* wsb = ws + (size_t)b * WS_STRIDE;
    const float  S   = wsb[1088];
    const float  c   = wsb[1089];

    float wcol[4];
    int   col[4];
    #pragma unroll
    for (int j = 0; j < 4; ++j) {
        col[j]  = colBase + wc * 64 + j * 16 + lrow;
        wcol[j] = wsb[col[j]];
    }

    #pragma unroll
    for (int i = 0; i < 2; ++i) {
        #pragma unroll
        for (int r = 0; r < 8; ++r) {
            int row = rowBase + wr * 32 + i * 16 + r + ((lane >> 4) << 3);
            float base = wsb[row] + c;
            bool  m    = mask[(size_t)b * N_ + row] != 0;
            float* po  = out + (size_t)b * ((size_t)N_ * N_) + (size_t)row * N_;
            #pragma unroll
            for (int j = 0; j < 4; ++j) {
                float val = S * acc[i][j][r] + base + wcol[j];
                po[col[j]] = m ? val : 0.f;
            }
        }
    }
}

// ---------------------------------------------------------------------------
extern "C" void kernel_launch(void* const* d_in, const int* in_sizes, int n_in,
                              void* d_out, int out_size, void* d_ws, size_t ws_size,
                              hipStream_t stream) {
    const float*         first  = (const float*)d_in[0];
    const float*         second = (const float*)d_in[1];
    const float*         classc = (const float*)d_in[2];
    const int*           y      = (const int*)d_in[3];
    const unsigned char* maskp  = (const unsigned char*)d_in[4];  // bool array
    const float*         W1     = (const float*)d_in[5];
    const float*         b1     = (const float*)d_in[6];
    float*               out    = (float*)d_out;
    float*               ws     = (float*)d_ws;  // needs 16*1152*4 = 73728 B

    caps_prep_kernel<<<B_, 256, 0, stream>>>(first, second, classc, y, W1, b1, ws);
    caps_gram_kernel<<<dim3(N_ / TILE, N_ / TILE, B_), 256, 0, stream>>>(first, maskp, ws, out);
}